// Decoder_86526411145547
// MI455X (gfx1250) — compile-verified
//
#include <hip/hip_runtime.h>
#include <hip/hip_bf16.h>

typedef float v2f __attribute__((ext_vector_type(2)));
typedef float v4f __attribute__((ext_vector_type(4)));
typedef float v8f __attribute__((ext_vector_type(8)));

#define NEG_HALF_LOG2E (-0.7213475204444817f)   // -0.5 * log2(e)

// ---------------------------------------------------------------------------
// Phase 1: per-gaussian quadratic-form coefficients, pre-scaled and
// pre-swizzled for the WMMA B-fragment layout.
//   Sinv = inv(cov) (adjugate), b = Sinv*mu, c = mu.b
//   logical psi[0..15] = {S00,S11,S22,S33, S01+S10,S02+S20,S03+S30,
//                         S12+S21,S13+S31,S23+S32, -2b0..-2b3, c, 0} * (-0.5*log2e)
//   stored order: half0 K-slots {0,1,4,5,8,9,12,13}, half1 {2,3,6,7,10,11,14,15}
//   so each lane reads its 32 bytes with two global_load_b128.
// ---------------------------------------------------------------------------
__global__ void psi_kernel(const float* __restrict__ means,
                           const float* __restrict__ covs,
                           float* __restrict__ psi, int M)
{
    int m = blockIdx.x * blockDim.x + threadIdx.x;
    if (m >= M) return;
    const float* a = covs + m * 16;
    float inv[16];
    inv[0]  =  a[5]*a[10]*a[15] - a[5]*a[11]*a[14] - a[9]*a[6]*a[15] + a[9]*a[7]*a[14] + a[13]*a[6]*a[11] - a[13]*a[7]*a[10];
    inv[4]  = -a[4]*a[10]*a[15] + a[4]*a[11]*a[14] + a[8]*a[6]*a[15] - a[8]*a[7]*a[14] - a[12]*a[6]*a[11] + a[12]*a[7]*a[10];
    inv[8]  =  a[4]*a[9]*a[15]  - a[4]*a[11]*a[13] - a[8]*a[5]*a[15] + a[8]*a[7]*a[13] + a[12]*a[5]*a[11] - a[12]*a[7]*a[9];
    inv[12] = -a[4]*a[9]*a[14]  + a[4]*a[10]*a[13] + a[8]*a[5]*a[14] - a[8]*a[6]*a[13] - a[12]*a[5]*a[10] + a[12]*a[6]*a[9];
    inv[1]  = -a[1]*a[10]*a[15] + a[1]*a[11]*a[14] + a[9]*a[2]*a[15] - a[9]*a[3]*a[14] - a[13]*a[2]*a[11] + a[13]*a[3]*a[10];
    inv[5]  =  a[0]*a[10]*a[15] - a[0]*a[11]*a[14] - a[8]*a[2]*a[15] + a[8]*a[3]*a[14] + a[12]*a[2]*a[11] - a[12]*a[3]*a[10];
    inv[9]  = -a[0]*a[9]*a[15]  + a[0]*a[11]*a[13] + a[8]*a[1]*a[15] - a[8]*a[3]*a[13] - a[12]*a[1]*a[11] + a[12]*a[3]*a[9];
    inv[13] =  a[0]*a[9]*a[14]  - a[0]*a[10]*a[13] - a[8]*a[1]*a[14] + a[8]*a[2]*a[13] + a[12]*a[1]*a[10] - a[12]*a[2]*a[9];
    inv[2]  =  a[1]*a[6]*a[15]  - a[1]*a[7]*a[14]  - a[5]*a[2]*a[15] + a[5]*a[3]*a[14] + a[13]*a[2]*a[7]  - a[13]*a[3]*a[6];
    inv[6]  = -a[0]*a[6]*a[15]  + a[0]*a[7]*a[14]  + a[4]*a[2]*a[15] - a[4]*a[3]*a[14] - a[12]*a[2]*a[7]  + a[12]*a[3]*a[6];
    inv[10] =  a[0]*a[5]*a[15]  - a[0]*a[7]*a[13]  - a[4]*a[1]*a[15] + a[4]*a[3]*a[13] + a[12]*a[1]*a[7]  - a[12]*a[3]*a[5];
    inv[14] = -a[0]*a[5]*a[14]  + a[0]*a[6]*a[13]  + a[4]*a[1]*a[14] - a[4]*a[2]*a[13] - a[12]*a[1]*a[6]  + a[12]*a[2]*a[5];
    inv[3]  = -a[1]*a[6]*a[11]  + a[1]*a[7]*a[10]  + a[5]*a[2]*a[11] - a[5]*a[3]*a[10] - a[9]*a[2]*a[7]   + a[9]*a[3]*a[6];
    inv[7]  =  a[0]*a[6]*a[11]  - a[0]*a[7]*a[10]  - a[4]*a[2]*a[11] + a[4]*a[3]*a[10] + a[8]*a[2]*a[7]   - a[8]*a[3]*a[6];
    inv[11] = -a[0]*a[5]*a[11]  + a[0]*a[7]*a[9]   + a[4]*a[1]*a[11] - a[4]*a[3]*a[9]  - a[8]*a[1]*a[7]   + a[8]*a[3]*a[5];
    inv[15] =  a[0]*a[5]*a[10]  - a[0]*a[6]*a[9]   - a[4]*a[1]*a[10] + a[4]*a[2]*a[9]  + a[8]*a[1]*a[6]   - a[8]*a[2]*a[5];
    float det  = a[0]*inv[0] + a[1]*inv[4] + a[2]*inv[8] + a[3]*inv[12];
    float rdet = 1.0f / det;
#pragma unroll
    for (int i = 0; i < 16; i++) inv[i] *= rdet;

    const float mu0 = means[m*4+0], mu1 = means[m*4+1];
    const float mu2 = means[m*4+2], mu3 = means[m*4+3];
    const float b0 = inv[0]*mu0 + inv[1]*mu1 + inv[2]*mu2  + inv[3]*mu3;
    const float b1 = inv[4]*mu0 + inv[5]*mu1 + inv[6]*mu2  + inv[7]*mu3;
    const float b2 = inv[8]*mu0 + inv[9]*mu1 + inv[10]*mu2 + inv[11]*mu3;
    const float b3 = inv[12]*mu0+ inv[13]*mu1+ inv[14]*mu2 + inv[15]*mu3;
    const float cc = mu0*b0 + mu1*b1 + mu2*b2 + mu3*b3;

    float q[16];
    q[0]  = inv[0];          q[1]  = inv[5];          q[2] = inv[10];          q[3] = inv[15];
    q[4]  = inv[1]+inv[4];   q[5]  = inv[2]+inv[8];   q[6] = inv[3]+inv[12];
    q[7]  = inv[6]+inv[9];   q[8]  = inv[7]+inv[13];  q[9] = inv[11]+inv[14];
    q[10] = -2.0f*b0;        q[11] = -2.0f*b1;        q[12] = -2.0f*b2;        q[13] = -2.0f*b3;
    q[14] = cc;              q[15] = 0.0f;
#pragma unroll
    for (int i = 0; i < 16; i++) q[i] *= NEG_HALF_LOG2E;   // fold -0.5 and log2e

    constexpr int perm[16] = {0,1,4,5,8,9,12,13, 2,3,6,7,10,11,14,15};
    float* o = psi + m * 16;
#pragma unroll
    for (int i = 0; i < 16; i++) o[i] = q[perm[i]];
}

// ---------------------------------------------------------------------------
// Phase 2: rank-16 FP32 GEMM via V_WMMA_F32_16X16X4_F32 (chained K=4 x4).
// Each wave owns TWO 16-ray A tiles (32 rays) so every B fragment + label
// load feeds 8 WMMAs. Epilogue: raw v_exp_f32 (all scales pre-folded into
// psi), label FMA, cross-lane row reduction, sigmoid.
// ---------------------------------------------------------------------------
__device__ __forceinline__ void make_a_frag(const float* __restrict__ origins,
                                            const float* __restrict__ dirs,
                                            int ray, int half, v2f a[4])
{
    const float p0 = origins[2*ray+0], p1 = origins[2*ray+1];
    const float p2 = dirs[2*ray+0],    p3 = dirs[2*ray+1];
    float phi[16];
    phi[0]=p0*p0; phi[1]=p1*p1; phi[2]=p2*p2; phi[3]=p3*p3;
    phi[4]=p0*p1; phi[5]=p0*p2; phi[6]=p0*p3;
    phi[7]=p1*p2; phi[8]=p1*p3; phi[9]=p2*p3;
    phi[10]=p0; phi[11]=p1; phi[12]=p2; phi[13]=p3;
    phi[14]=1.0f; phi[15]=0.0f;
#pragma unroll
    for (int i = 0; i < 4; i++) {             // constant indices -> cndmask selects
        a[i].x = half ? phi[4*i+2] : phi[4*i+0];
        a[i].y = half ? phi[4*i+3] : phi[4*i+1];
    }
}

__global__ __launch_bounds__(256) void decoder_kernel(
    const float* __restrict__ origins, const float* __restrict__ dirs,
    const float* __restrict__ psi, const float* __restrict__ labels,
    float* __restrict__ out, int N, int M)
{
    const int lane    = threadIdx.x & 31;
    const int wave    = blockIdx.x * (blockDim.x >> 5) + (threadIdx.x >> 5);
    const int rayBase = wave << 5;            // 32 rays per wave
    if (rayBase >= N) return;                 // uniform per wave: EXEC stays all-1s
    const int row  = lane & 15;
    const int half = lane >> 4;

    v2f aA[4], aB[4];
    make_a_frag(origins, dirs, rayBase + row,      half, aA);
    make_a_frag(origins, dirs, rayBase + 16 + row, half, aB);

    float accA[8] = {0.f,0.f,0.f,0.f,0.f,0.f,0.f,0.f};
    float accB[8] = {0.f,0.f,0.f,0.f,0.f,0.f,0.f,0.f};

    for (int t = 0; t < M; t += 16) {
        const int m = t + row;                       // this lane's gaussian column
        const float* pb = psi + (m << 4) + (half << 3);
        const v4f q0 = *(const v4f*)(pb + 0);        // global_load_b128
        const v4f q1 = *(const v4f*)(pb + 4);        // global_load_b128
        const v2f b0 = __builtin_shufflevector(q0, q0, 0, 1);
        const v2f b1 = __builtin_shufflevector(q0, q0, 2, 3);
        const v2f b2 = __builtin_shufflevector(q1, q1, 0, 1);
        const v2f b3 = __builtin_shufflevector(q1, q1, 2, 3);
        const float lab = labels[m];

        v8f cA = {0.f,0.f,0.f,0.f,0.f,0.f,0.f,0.f};
        v8f cB = {0.f,0.f,0.f,0.f,0.f,0.f,0.f,0.f};
        cA = __builtin_amdgcn_wmma_f32_16x16x4_f32(false, aA[0], false, b0, (short)0, cA, false, false);
        cB = __builtin_amdgcn_wmma_f32_16x16x4_f32(false, aB[0], false, b0, (short)0, cB, false, false);
        cA = __builtin_amdgcn_wmma_f32_16x16x4_f32(false, aA[1], false, b1, (short)0, cA, false, false);
        cB = __builtin_amdgcn_wmma_f32_16x16x4_f32(false, aB[1], false, b1, (short)0, cB, false, false);
        cA = __builtin_amdgcn_wmma_f32_16x16x4_f32(false, aA[2], false, b2, (short)0, cA, false, false);
        cB = __builtin_amdgcn_wmma_f32_16x16x4_f32(false, aB[2], false, b2, (short)0, cB, false, false);
        cA = __builtin_amdgcn_wmma_f32_16x16x4_f32(false, aA[3], false, b3, (short)0, cA, false, false);
        cB = __builtin_amdgcn_wmma_f32_16x16x4_f32(false, aB[3], false, b3, (short)0, cB, false, false);

#pragma unroll
        for (int v = 0; v < 8; v++) {
            // raw v_exp_f32: arguments below -126 only occur when the true
            // contribution is < 2^-126 ~ 0, so flush-to-zero is exact enough.
            accA[v] += __builtin_amdgcn_exp2f(cA[v]) * lab;
            accB[v] += __builtin_amdgcn_exp2f(cB[v]) * lab;
        }
    }

    // Reduce over the 16 lanes of each half (gaussian columns) -> per-ray sums
#pragma unroll
    for (int v = 0; v < 8; v++) {
        float sA = accA[v], sB = accB[v];
        sA += __shfl_xor(sA, 1);  sB += __shfl_xor(sB, 1);
        sA += __shfl_xor(sA, 2);  sB += __shfl_xor(sB, 2);
        sA += __shfl_xor(sA, 4);  sB += __shfl_xor(sB, 4);
        sA += __shfl_xor(sA, 8);  sB += __shfl_xor(sB, 8);
        accA[v] = sA; accB[v] = sB;
    }
    if (row == 0) {
#pragma unroll
        for (int v = 0; v < 8; v++) {
            const int r = v + 8 * half;
            out[rayBase + r]      = 1.0f / (1.0f + __expf(-accA[v]));
            out[rayBase + 16 + r] = 1.0f / (1.0f + __expf(-accB[v]));
        }
    }
}

extern "C" void kernel_launch(void* const* d_in, const int* in_sizes, int n_in,
                              void* d_out, int out_size, void* d_ws, size_t ws_size,
                              hipStream_t stream) {
    const float* origins = (const float*)d_in[0];   // [N,2]
    const float* dirs    = (const float*)d_in[1];   // [N,2]
    const float* means   = (const float*)d_in[2];   // [M,4]
    const float* covs    = (const float*)d_in[3];   // [M,4,4]
    const float* labels  = (const float*)d_in[4];   // [M]
    float* out = (float*)d_out;
    float* psi = (float*)d_ws;                      // M*16 floats (64 KB)

    const int N = in_sizes[0] / 2;
    const int M = in_sizes[4];

    psi_kernel<<<(M + 255) / 256, 256, 0, stream>>>(means, covs, psi, M);

    const int waves  = N / 32;                       // one wave per 32 rays
    const int blocks = (waves + 7) / 8;              // 8 wave32s per 256-thread block
    decoder_kernel<<<blocks, 256, 0, stream>>>(origins, dirs, psi, labels, out, N, M);
}